// DEPLayer_1271310320207
// MI455X (gfx1250) — compile-verified
//
#include <hip/hip_runtime.h>
#include <hip/hip_bf16.h>

#define Bn 8
#define Tn 128
#define Dn 400
#define Hn 300
#define Ln 45

typedef __attribute__((ext_vector_type(2))) float v2f;
typedef __attribute__((ext_vector_type(8))) float v8f;

// ---------------------------------------------------------------------------
// Kernel 1: P_a = cont @ W1a, P_b = cont @ W1b via V_WMMA_F32_16X16X4_F32.
// cont is [B*T, D] = [1024, 400] row-major; W is [D, H] = [400, 300] row-major.
// One wave computes TWO adjacent 16x16 output tiles (cols n..n+15, n+16..n+31)
// sharing one A fragment -> two independent WMMA accumulation chains.
// grid = (ceil(H/32)=10, (B*T)/16=64, 2); blockIdx.z selects {W1a->Pa, W1b->Pb}.
// Out-of-range columns (300..319) load clamped-in-bounds garbage; since WMMA
// output columns are independent, garbage only lands in D columns we never
// store, so no masking math is needed in the inner loop.
// ---------------------------------------------------------------------------
__global__ __launch_bounds__(32)
void proj_gemm_wmma(const float* __restrict__ cont,
                    const float* __restrict__ W1a,
                    const float* __restrict__ W1b,
                    float* __restrict__ Pa,
                    float* __restrict__ Pb)
{
    const int lane = threadIdx.x;          // 0..31
    const int half = lane >> 4;            // 0 or 1
    const int l16  = lane & 15;
    const int mbase = blockIdx.y * 16;
    const int nbase = blockIdx.x * 32;

    const float* __restrict__ W = blockIdx.z ? W1b : W1a;
    float* __restrict__ P       = blockIdx.z ? Pb  : Pa;

    // A fragment (16x4 f32): lanes 0-15 row M=lane,   K = k..k+1;
    //                        lanes 16-31 row M=lane-16, K = k+2..k+3.
    const int arow = mbase + l16;
    // B fragment (4x16 f32): V0 = row (k + 2*half), V1 = row (k + 2*half + 1),
    //                        column N = l16 for both lane halves.
    const int col0 = nbase + l16;
    const int col1 = col0 + 16;
    const int c0 = (col0 < Hn) ? col0 : (Hn - 1);   // clamp (stores are guarded)
    const int c1 = (col1 < Hn) ? col1 : (Hn - 1);

    const float* __restrict__ ap = cont + (size_t)arow * Dn + half * 2;

    v8f acc0 = {};
    v8f acc1 = {};
    for (int k = 0; k < Dn; k += 4) {
        v2f a;
        a.x = ap[k];
        a.y = ap[k + 1];
        const int kb = k + half * 2;
        const float* __restrict__ w0 = W + (size_t)kb * Hn;
        const float* __restrict__ w1 = W + (size_t)(kb + 1) * Hn;
        v2f b0, b1;
        b0.x = w0[c0]; b0.y = w1[c0];
        b1.x = w0[c1]; b1.y = w1[c1];
        acc0 = __builtin_amdgcn_wmma_f32_16x16x4_f32(
            false, a, false, b0, (short)0, acc0, false, false);
        acc1 = __builtin_amdgcn_wmma_f32_16x16x4_f32(
            false, a, false, b1, (short)0, acc1, false, false);
    }

    // D layout: VGPR j -> row (mbase + j + 8*half), col = nbase + l16 (+16)
    if (col0 < Hn) {
        #pragma unroll
        for (int j = 0; j < 8; ++j) {
            const int row = mbase + j + half * 8;
            P[(size_t)row * Hn + col0] = acc0[j];
        }
    }
    if (col1 < Hn) {
        #pragma unroll
        for (int j = 0; j < 8; ++j) {
            const int row = mbase + j + half * 8;
            P[(size_t)row * Hn + col1] = acc1[j];
        }
    }
}

// ---------------------------------------------------------------------------
// Kernel 2: root_b[h] = root(1x400) @ W1b(400x300)  (shared by all batches)
// ---------------------------------------------------------------------------
__global__ void root_proj(const float* __restrict__ root,
                          const float* __restrict__ W1b,
                          float* __restrict__ rootb)
{
    const int h = blockIdx.x * blockDim.x + threadIdx.x;
    if (h < Hn) {
        float acc = 0.0f;
        for (int d = 0; d < Dn; ++d)
            acc = fmaf(root[d], W1b[(size_t)d * Hn + h], acc);
        rootb[h] = acc;
    }
}

// ---------------------------------------------------------------------------
// Kernel 3: one block per (b, t). Fuses relu(a+b+b1), arc logits over the
// 129 head candidates, softmax CE + argmax, selected-head gather, label
// logits over L=45, label CE. Writes per-token results (deterministic).
// ---------------------------------------------------------------------------
__global__ __launch_bounds__(128)
void arc_label_kernel(const float* __restrict__ Pa,
                      const float* __restrict__ Pb,
                      const float* __restrict__ rootb,
                      const float* __restrict__ b1,
                      const float* __restrict__ Wa,
                      const float* __restrict__ ba,
                      const float* __restrict__ Wl,
                      const float* __restrict__ bl,
                      const int*   __restrict__ slen,
                      const int*   __restrict__ darcs,
                      const int*   __restrict__ dlabs,
                      const int*   __restrict__ use_desired,
                      float* __restrict__ uas_out,
                      float* __restrict__ las_out,
                      float* __restrict__ msk_out)
{
    const int t = blockIdx.x;
    const int b = blockIdx.y;
    const int tid = threadIdx.x;

    __shared__ float sA[Hn + 4];     // a_proj[b,t,:] + b1
    __shared__ float sWa[Hn + 4];
    __shared__ float sR[Hn + 4];     // relu vector of selected head
    __shared__ float slog[Tn + 4];   // 129 arc logits
    __shared__ float lab[Ln + 3];
    __shared__ float sh_lse;
    __shared__ int   sh_arg;

    const float* __restrict__ pa = Pa + ((size_t)b * Tn + t) * Hn;
    for (int h = tid; h < Hn; h += blockDim.x) {
        sA[h]  = pa[h] + b1[h];
        sWa[h] = Wa[h];
    }
    __syncthreads();

    // Arc logits for s in [0, T] (s=0 is the root head candidate).
    const float ba0 = ba[0];
    for (int s = tid; s <= Tn; s += blockDim.x) {
        const float* __restrict__ pb =
            (s == 0) ? rootb : (Pb + ((size_t)b * Tn + (s - 1)) * Hn);
        float acc = 0.0f;
        for (int h = 0; h < Hn; ++h) {
            float v = sA[h] + pb[h];
            v = v > 0.0f ? v : 0.0f;
            acc = fmaf(v, sWa[h], acc);
        }
        slog[s] = acc + ba0;
    }
    __syncthreads();

    if (tid == 0) {
        float m = slog[0];
        int   am = 0;
        for (int s = 1; s <= Tn; ++s) {       // strict '>' => first-max, as jnp.argmax
            if (slog[s] > m) { m = slog[s]; am = s; }
        }
        float se = 0.0f;
        for (int s = 0; s <= Tn; ++s) se += expf(slog[s] - m);
        sh_lse = m + logf(se);
        sh_arg = am;
    }
    __syncthreads();

    const int da  = darcs[b * Tn + t];
    const int sel = use_desired[0] ? da : sh_arg;
    const float maskv = (t < slen[b]) ? 1.0f : 0.0f;

    // relu(a + b_sel + b1) for the selected head, cached in LDS
    const float* __restrict__ pbs =
        (sel == 0) ? rootb : (Pb + ((size_t)b * Tn + (sel - 1)) * Hn);
    for (int h = tid; h < Hn; h += blockDim.x) {
        float v = sA[h] + pbs[h];
        sR[h] = v > 0.0f ? v : 0.0f;
    }
    __syncthreads();

    // Label logits (L = 45)
    for (int l = tid; l < Ln; l += blockDim.x) {
        float acc = bl[l];
        for (int h = 0; h < Hn; ++h)
            acc = fmaf(sR[h], Wl[(size_t)h * Ln + l], acc);
        lab[l] = acc;
    }
    __syncthreads();

    if (tid == 0) {
        float m = lab[0];
        for (int l = 1; l < Ln; ++l) m = fmaxf(m, lab[l]);
        float se = 0.0f;
        for (int l = 0; l < Ln; ++l) se += expf(lab[l] - m);
        const float lab_lse = m + logf(se);

        const int dl = dlabs[b * Tn + t];
        const int tok = b * Tn + t;
        uas_out[tok] = (sh_lse - slog[da]) * maskv;
        las_out[tok] = (lab_lse - lab[dl]) * maskv;
        msk_out[tok] = maskv;
    }
}

// ---------------------------------------------------------------------------
// Kernel 4: fixed-order reduction -> scalar loss (deterministic).
// ---------------------------------------------------------------------------
__global__ void finalize_kernel(const float* __restrict__ uas,
                                const float* __restrict__ las,
                                const float* __restrict__ msk,
                                float* __restrict__ out)
{
    if (threadIdx.x == 0 && blockIdx.x == 0) {
        float su = 0.0f, sl = 0.0f, sm = 0.0f;
        for (int i = 0; i < Bn * Tn; ++i) {
            su += uas[i];
            sl += las[i];
            sm += msk[i];
        }
        const float nv = sm > 1.0f ? sm : 1.0f;
        out[0] = (su + sl) / (2.0f * nv);
    }
}

extern "C" void kernel_launch(void* const* d_in, const int* in_sizes, int n_in,
                              void* d_out, int out_size, void* d_ws, size_t ws_size,
                              hipStream_t stream)
{
    (void)in_sizes; (void)n_in; (void)out_size; (void)ws_size;

    const float* cont = (const float*)d_in[0];
    const float* root = (const float*)d_in[1];
    const float* W1a  = (const float*)d_in[2];
    const float* W1b  = (const float*)d_in[3];
    const float* b1   = (const float*)d_in[4];
    const float* Wa   = (const float*)d_in[5];
    const float* ba   = (const float*)d_in[6];
    const float* Wl   = (const float*)d_in[7];
    const float* bl   = (const float*)d_in[8];
    const int* slen   = (const int*)d_in[9];
    const int* darcs  = (const int*)d_in[10];
    const int* dlabs  = (const int*)d_in[11];
    const int* usedes = (const int*)d_in[12];
    float* out = (float*)d_out;

    float* ws    = (float*)d_ws;
    float* Pa    = ws;                                 // [1024, 300]
    float* Pb    = Pa + (size_t)Bn * Tn * Hn;          // [1024, 300]
    float* rootb = Pb + (size_t)Bn * Tn * Hn;          // [300] (+pad)
    float* uasv  = rootb + 320;                        // [1024]
    float* lasv  = uasv + Bn * Tn;                     // [1024]
    float* mskv  = lasv + Bn * Tn;                     // [1024]

    dim3 ggrid((Hn + 31) / 32, (Bn * Tn) / 16, 2);     // (10, 64, 2)
    proj_gemm_wmma<<<ggrid, 32, 0, stream>>>(cont, W1a, W1b, Pa, Pb);

    root_proj<<<1, 320, 0, stream>>>(root, W1b, rootb);

    dim3 agrid(Tn, Bn);                                // (128, 8)
    arc_label_kernel<<<agrid, 128, 0, stream>>>(Pa, Pb, rootb, b1, Wa, ba, Wl, bl,
                                                slen, darcs, dlabs, usedes,
                                                uasv, lasv, mskv);

    finalize_kernel<<<1, 32, 0, stream>>>(uasv, lasv, mskv, out);
}